// CUDAFusedRejectionSampler_22445499089619
// MI455X (gfx1250) — compile-verified
//
#include <hip/hip_runtime.h>
#include <cstdint>

// Problem constants (match the reference).
#define B_ 64
#define K_ 8
#define V_ 128000
#define PLACEHOLDER_ (-1)

#define NTHREADS 256
#define NWAVES   (NTHREADS / 32)
#define PER_THREAD 20                        // floats per thread per chunk (5 x b128 per row)
#define CHUNK (NTHREADS * PER_THREAD)        // 5120 floats
#define NCHUNK (V_ / CHUNK)                  // 25, exact (no tail)
#define NPER (2 * PER_THREAD / 4)            // 10 async b128 ops per thread per chunk

// ---- CDNA5 async global -> LDS staging (gfx1250) ------------------------------
// LDS address is the vdst VGPR plus the instruction offset; low 32 bits of a flat shared pointer
// are the wave-logical LDS byte address.
__device__ __forceinline__ void async_b128_to_lds(uint32_t lds_byte_addr,
                                                  const float* gaddr) {
    asm volatile("global_load_async_to_lds_b128 %0, %1, off"
                 :
                 : "v"(lds_byte_addr), "v"(gaddr)
                 : "memory");
}

// Wave-local: drain pending DS reads of the buffer we are about to overwrite,
// then issue the 10 async b128 loads for one chunk of both rows.
__device__ __forceinline__ void stage_chunk(const float* __restrict__ rowT,
                                            const float* __restrict__ rowD,
                                            int base, int tbase,
                                            float* bufT, float* bufD) {
    asm volatile("s_wait_dscnt 0" ::: "memory");
    const uint32_t lT = (uint32_t)(uintptr_t)&bufT[tbase];
    const uint32_t lD = (uint32_t)(uintptr_t)&bufD[tbase];
    const float* gT = rowT + base + tbase;
    const float* gD = rowD + base + tbase;
#pragma unroll
    for (int i = 0; i < PER_THREAD; i += 4) {
        async_b128_to_lds(lT + (uint32_t)(i * 4), gT + i);
        async_b128_to_lds(lD + (uint32_t)(i * 4), gD + i);
    }
}

__device__ __forceinline__ void wait_async_pipe(bool pipelined) {
    if (pipelined) asm volatile("s_wait_asynccnt 10" ::: "memory");  // NPER newest still in flight
    else           asm volatile("s_wait_asynccnt 0"  ::: "memory");
}

__global__ __launch_bounds__(NTHREADS) void rejection_sample_kernel(
    const int*   __restrict__ draft_ids,     // [B,K]
    const float* __restrict__ draft_probs,   // [B,K,V]
    const float* __restrict__ target_probs,  // [B,K,V]
    const int*   __restrict__ bonus_ids,     // [B]
    const float* __restrict__ uniform,       // [B,K]
    int*         __restrict__ out)           // 832 int32
{
    const int b    = blockIdx.x;
    const int t    = threadIdx.x;
    const int lane = t & 31;
    const int wave = t >> 5;

    __shared__ __align__(16) float sT[2][CHUNK];   // 40 KB
    __shared__ __align__(16) float sD[2][CHUNK];   // 40 KB
    __shared__ float waveSum[NCHUNK * NWAVES];
    __shared__ float chunkSum[NCHUNK];
    __shared__ float red[NTHREADS];
    __shared__ float sBV[NTHREADS];
    __shared__ int   sBI[NTHREADS];
    __shared__ int   s_j, s_num_acc, s_cstar, s_fallback, s_minIdx;
    __shared__ float s_thr, s_prefix;

    // ---- Phase A: accept chain, one gather per lane, ballot + ctz -----------
    if (t < 32) {
        bool acc = false;
        if (t < K_) {
            const int    tok = draft_ids[b * K_ + t];
            const size_t row = (size_t)(b * K_ + t) * (size_t)V_;
            const float  p   = target_probs[row + tok];
            const float  q   = draft_probs[row + tok];
            const float  ap  = fminf(1.0f, p / fmaxf(q, 1e-10f));
            acc = uniform[b * K_ + t] < ap;
        }
        const unsigned mask = (unsigned)__ballot(acc);
        if (t == 0) {
            const unsigned rej = (~mask) & 0xFFu;          // first 0 among k=0..7
            const int num_acc  = rej ? (int)__builtin_ctz(rej) : K_;
            s_num_acc = num_acc;
            s_j       = (num_acc < K_) ? num_acc : (K_ - 1);
            s_minIdx  = 0x7fffffff;
        }
    }
    __syncthreads();

    const int    j     = s_j;
    const float* rowT  = target_probs + (size_t)(b * K_ + j) * (size_t)V_;
    const float* rowD  = draft_probs  + (size_t)(b * K_ + j) * (size_t)V_;
    const int    tbase = t * PER_THREAD;

    // ---- Pass 1: barrier-free double-buffered async stream ------------------
    // Each thread consumes only the LDS region its own async ops staged.
    stage_chunk(rowT, rowD, 0, tbase, sT[0], sD[0]);
    for (int c = 0; c < NCHUNK; ++c) {
        const bool pipe = (c + 1) < NCHUNK;
        if (pipe)
            stage_chunk(rowT, rowD, (c + 1) * CHUNK, tbase, sT[(c + 1) & 1], sD[(c + 1) & 1]);
        wait_async_pipe(pipe);

        const float* bT = sT[c & 1];
        const float* bD = sD[c & 1];
        float part = 0.0f;
#pragma unroll
        for (int i = 0; i < PER_THREAD; ++i)
            part += fmaxf(bT[tbase + i] - bD[tbase + i], 0.0f);

        // wave32 shuffle reduce -> lane 0
#pragma unroll
        for (int off = 16; off > 0; off >>= 1)
            part += __shfl_down(part, off, 32);
        if (lane == 0) waveSum[c * NWAVES + wave] = part;
    }
    __syncthreads();

    // ---- Chunk sums (25 threads), then thread 0: total / threshold / chunk --
    if (t < NCHUNK) {
        float s = 0.0f;
        for (int w = 0; w < NWAVES; ++w) s += waveSum[t * NWAVES + w];
        chunkSum[t] = s;
    }
    __syncthreads();
    if (t == 0) {
        float total = 0.0f;
        for (int c = 0; c < NCHUNK; ++c) total += chunkSum[c];
        const float thr = uniform[b * K_ + j] * total;

        int   cstar  = -1;    // -1: no crossing (argmax over all-False == 0)
        float prefix = 0.0f;
        float run    = 0.0f;
        for (int c = 0; c < NCHUNK; ++c) {
            const float nr = run + chunkSum[c];
            if (nr > thr) { cstar = c; prefix = run; break; }
            run = nr;
        }
        s_cstar  = (total > 0.0f) ? cstar : -2;   // -2: argmax(target[j]) fallback
        s_thr    = thr;
        s_prefix = prefix;
    }
    __syncthreads();

    int recovered = 0;
    const int cstar = s_cstar;

    // ---- Pass 2a (common): scan only the crossing chunk ---------------------
    if (cstar >= 0) {
        const int base = cstar * CHUNK;
        stage_chunk(rowT, rowD, base, tbase, sT[0], sD[0]);
        wait_async_pipe(false);

        float part = 0.0f;
#pragma unroll
        for (int i = 0; i < PER_THREAD; ++i)
            part += fmaxf(sT[0][tbase + i] - sD[0][tbase + i], 0.0f);
        red[t] = part;
        __syncthreads();
        if (t == 0) {   // serial exclusive scan keeps left-to-right order
            float run = s_prefix;
            for (int i = 0; i < NTHREADS; ++i) { const float v = red[i]; red[i] = run; run += v; }
        }
        __syncthreads();
        {
            float       run = red[t];
            const float thr = s_thr;
            for (int i = 0; i < PER_THREAD; ++i) {
                run += fmaxf(sT[0][tbase + i] - sD[0][tbase + i], 0.0f);
                if (run > thr) { atomicMin(&s_minIdx, base + tbase + i); break; }
            }
        }
        __syncthreads();
        recovered = (s_minIdx == 0x7fffffff) ? 0 : s_minIdx;
    }
    // ---- Pass 2b (rare): total <= 0 -> argmax(target[j]) --------------------
    else if (cstar == -2) {
        float bv = -1.0f; int bi = 0;
        for (int i = t; i < V_; i += NTHREADS) {
            const float v = rowT[i];
            if (v > bv) { bv = v; bi = i; }
        }
        sBV[t] = bv; sBI[t] = bi;
        __syncthreads();
        if (t == 0) {
            float rv = sBV[0]; int ri = sBI[0];
            for (int i = 1; i < NTHREADS; ++i) {
                if (sBV[i] > rv || (sBV[i] == rv && sBI[i] < ri)) { rv = sBV[i]; ri = sBI[i]; }
            }
            s_fallback = ri;
        }
        __syncthreads();
        recovered = s_fallback;
    }

    // ---- Outputs (thread 0) -------------------------------------------------
    if (t == 0) {
        const int num_acc = s_num_acc;
        const int all_acc = (num_acc == K_) ? 1 : 0;
        const int last    = all_acc ? bonus_ids[b] : recovered;
        int* orow = out + b * (K_ + 1);
        for (int pos = 0; pos <= K_; ++pos) {
            int v;
            if      (pos <  num_acc) v = draft_ids[b * K_ + pos];
            else if (pos == num_acc) v = last;
            else                     v = PLACEHOLDER_;
            orow[pos] = v;
        }
        const int off = B_ * (K_ + 1);
        out[off + 0 * B_ + b] = num_acc;        // num_accepted
        out[off + 1 * B_ + b] = num_acc;        // accepted_counts
        out[off + 2 * B_ + b] = 1 - all_acc;    // recovered_counts
        out[off + 3 * B_ + b] = all_acc;        // bonus_counts
    }
}

extern "C" void kernel_launch(void* const* d_in, const int* in_sizes, int n_in,
                              void* d_out, int out_size, void* d_ws, size_t ws_size,
                              hipStream_t stream) {
    (void)in_sizes; (void)n_in; (void)out_size; (void)d_ws; (void)ws_size;
    const int*   draft_ids    = (const int*)  d_in[0];  // [B,K] int32
    const float* draft_probs  = (const float*)d_in[1];  // [B,K,V] f32
    const float* target_probs = (const float*)d_in[2];  // [B,K,V] f32
    const int*   bonus_ids    = (const int*)  d_in[3];  // [B] int32
    const float* uniform      = (const float*)d_in[4];  // [B,K] f32
    int*         out          = (int*)d_out;            // 832 int32

    rejection_sample_kernel<<<dim3(B_), dim3(NTHREADS), 0, stream>>>(
        draft_ids, draft_probs, target_probs, bonus_ids, uniform, out);
}